// BilinearInteraction_44530220925268
// MI455X (gfx1250) — compile-verified
//
#include <hip/hip_runtime.h>

// BilinearInteraction for MI455X (gfx1250, wave32, WMMA).
// Y[b,p,f] = (sum_e x[b,i_p,e] * W[p,e,f]) * x[b,j_p,f]
// B=4096, F=24, E=64, P=276. fp32 throughout via V_WMMA_F32_16X16X4_F32:
// output stores (289 MB) dominate HBM traffic (~14us floor @ 23.3 TB/s),
// fp32 matrix rate is roughly balanced against that, so no precision cut.

typedef __attribute__((ext_vector_type(2))) float v2f;
typedef __attribute__((ext_vector_type(8))) float v8f;

#define FIELD_NUM 24
#define EDIM      64
#define PAIRS     276          // 24*23/2

// LDS layout for W[p]: pair-interleaved.  element (e, f) lives at dword
//   (e>>1)*PAIR_STRIDE + 2*f + (e&1)
// so a B-fragment {W[e][n], W[e+1][n]} is ONE aligned ds_load_b64.
// PAIR_STRIDE = 128 data + 32 pad dwords; 160 mod 64 = 32 => the half=1
// half-wave reads land 32 banks away from half=0 -> conflict-free b64 reads.
#define PAIR_STRIDE 160

__global__ __launch_bounds__(256) void
bilinear_wmma_f32(const float* __restrict__ x,
                  const float* __restrict__ W,
                  float* __restrict__ out) {
  __shared__ float wlds[(EDIM / 2) * PAIR_STRIDE];   // 20.5 KB

  const int p = blockIdx.y;
  // Decode linear pair index -> (i, j), i<j  (scalar, uniform per block)
  int i = 0, rem = p, span = FIELD_NUM - 1;
  while (rem >= span) { rem -= span; ++i; --span; }
  const int j = i + 1 + rem;

  const int tid = threadIdx.x;

  // ---- Stage W[p] (64x64 f32 = 16 KB) into pair-interleaved LDS ----
  // 32 e-pairs x 16 f-quads = 512 work items over 256 threads.
  {
    const float* wg = W + (size_t)p * EDIM * EDIM;
    #pragma unroll
    for (int it = 0; it < 2; ++it) {
      int w  = tid + it * 256;
      int pr = w >> 4;                 // e-pair index 0..31
      int fq = w & 15;                 // f-quad index 0..15
      float4 g0 = *(const float4*)(wg + (size_t)(2 * pr)     * EDIM + fq * 4);
      float4 g1 = *(const float4*)(wg + (size_t)(2 * pr + 1) * EDIM + fq * 4);
      float* dst = &wlds[pr * PAIR_STRIDE + fq * 8];
      *(v2f*)(dst + 0) = v2f{g0.x, g1.x};
      *(v2f*)(dst + 2) = v2f{g0.y, g1.y};
      *(v2f*)(dst + 4) = v2f{g0.z, g1.z};
      *(v2f*)(dst + 6) = v2f{g0.w, g1.w};
    }
  }
  __syncthreads();

  const int lane   = tid & 31;         // wave32
  const int wave   = tid >> 5;         // 0..7
  const int half   = lane >> 4;        // K-split half of the wave
  const int lane16 = lane & 15;
  const int b0     = blockIdx.x * 128 + wave * 16;   // 16 batch rows per wave

  // ---- A fragments: x[b0+m, i, :]  (16x4 f32 per k-chunk) ----
  // ISA layout: lanes 0-15 hold M=lane, K={0,1}; lanes 16-31 hold M=lane-16,
  // K={2,3}.  => per k-chunk, lane reads x[m][4*kc + 2*half + {0,1}] (b64).
  const float* xa = x + ((size_t)(b0 + lane16) * FIELD_NUM + i) * EDIM + 2 * half;
  v2f afrag[16];
  #pragma unroll
  for (int kc = 0; kc < 16; ++kc)
    afrag[kc] = *(const v2f*)(xa + kc * 4);

  v8f acc[4] = {v8f{}, v8f{}, v8f{}, v8f{}};   // 16x64 output tile

  // ---- 64 x v_wmma_f32_16x16x4_f32, B-fragments as single ds_load_b64 ----
  #pragma unroll
  for (int kc = 0; kc < 16; ++kc) {
    // B layout (4x16 KxN, row striped across lanes): lane needs
    // {W[e0][n], W[e0+1][n]}, e0 = 4*kc + 2*half, n = lane16 (+16*nt).
    const float* brow = &wlds[(kc * 2 + half) * PAIR_STRIDE + lane16 * 2];
    #pragma unroll
    for (int nt = 0; nt < 4; ++nt) {
      v2f bfrag = *(const v2f*)(brow + nt * 32);
      // 8 args: (neg_a, A, neg_b, B, c_mod, C, reuse_a, reuse_b)
      acc[nt] = __builtin_amdgcn_wmma_f32_16x16x4_f32(
          false, afrag[kc], false, bfrag, (short)0, acc[nt], false, false);
    }
  }

  // ---- Epilogue: multiply by x[:, j, :] per C/D layout, store coalesced ----
  // C/D 16x16 f32: VGPR r holds row m = r + 8*half, col n = lane16.
  const float* xj = x + ((size_t)(b0 + 8 * half) * FIELD_NUM + j) * EDIM + lane16;
  float*       op = out + ((size_t)(b0 + 8 * half) * PAIRS + p) * EDIM + lane16;
  #pragma unroll
  for (int nt = 0; nt < 4; ++nt) {
    #pragma unroll
    for (int r = 0; r < 8; ++r) {
      float xv = xj[(size_t)r * FIELD_NUM * EDIM + nt * 16];
      op[(size_t)r * PAIRS * EDIM + nt * 16] = acc[nt][r] * xv;
    }
  }
}

extern "C" void kernel_launch(void* const* d_in, const int* in_sizes, int n_in,
                              void* d_out, int out_size, void* d_ws, size_t ws_size,
                              hipStream_t stream) {
  const float* x = (const float*)d_in[0];   // (4096, 24, 64) f32
  const float* W = (const float*)d_in[1];   // (276, 64, 64) f32
  float* out = (float*)d_out;               // (4096, 276, 64) f32

  const int Btotal = in_sizes[0] / (FIELD_NUM * EDIM);   // 4096
  dim3 grid(Btotal / 128, PAIRS);                        // (32, 276)
  bilinear_wmma_f32<<<grid, dim3(256), 0, stream>>>(x, W, out);
}